// ResGraphConvPool_73890617360432
// MI455X (gfx1250) — compile-verified
//
#include <hip/hip_runtime.h>
#include <hip/hip_bf16.h>

typedef __attribute__((ext_vector_type(2))) float v2f;
typedef __attribute__((ext_vector_type(8))) float v8f;

#define NBATCH 8
#define NPTS   4096
#define DFEAT  64
#define KNN    8
#define NBLK   12
#define BN_EPS 1e-5f
#define SLOPE  0.01f
#define INV9   (1.0f / 9.0f)

// ---------------------------------------------------------------------------
// KNN: for each (b, q) find 8 nearest neighbors (incl. self), ascending dist.
// ---------------------------------------------------------------------------
__global__ __launch_bounds__(256) void knn_kernel(const float* __restrict__ xyz,
                                                  int* __restrict__ idx) {
  __shared__ float cx[256], cy[256], cz[256], cs[256];
  const int b = blockIdx.y;
  const int q = blockIdx.x * 256 + threadIdx.x;
  const float* xb = xyz + (size_t)b * NPTS * 3;
  const float qx = xb[q * 3 + 0];
  const float qy = xb[q * 3 + 1];
  const float qz = xb[q * 3 + 2];
  const float qs = qx * qx + qy * qy + qz * qz;

  float bd[KNN];
  int   bi[KNN];
#pragma unroll
  for (int j = 0; j < KNN; ++j) { bd[j] = 3.0e38f; bi[j] = 0; }

  for (int m0 = 0; m0 < NPTS; m0 += 256) {
    __syncthreads();
    {
      const int m = m0 + threadIdx.x;
      const float x = xb[m * 3 + 0];
      const float y = xb[m * 3 + 1];
      const float z = xb[m * 3 + 2];
      cx[threadIdx.x] = x; cy[threadIdx.x] = y; cz[threadIdx.x] = z;
      cs[threadIdx.x] = x * x + y * y + z * z;
    }
    __syncthreads();
    for (int t = 0; t < 256; ++t) {
      const float d = qs + cs[t] - 2.0f * (qx * cx[t] + qy * cy[t] + qz * cz[t]);
      if (d < bd[KNN - 1]) {
        bd[KNN - 1] = d; bi[KNN - 1] = m0 + t;
#pragma unroll
        for (int s = KNN - 1; s > 0; --s) {
          if (bd[s] < bd[s - 1]) {
            float td = bd[s]; bd[s] = bd[s - 1]; bd[s - 1] = td;
            int   ti = bi[s]; bi[s] = bi[s - 1]; bi[s - 1] = ti;
          }
        }
      }
    }
  }
  int* ib = idx + ((size_t)b * NPTS + q) * KNN;
#pragma unroll
  for (int j = 0; j < KNN; ++j) ib[j] = bi[j];
}

// ---------------------------------------------------------------------------
// Block-0 activation + transpose: xT[b][n][d] = leaky(bn0(points[b][d][n]))
// ---------------------------------------------------------------------------
__global__ __launch_bounds__(256) void bnact_transpose_kernel(
    const float* __restrict__ pts, const float* __restrict__ gamma,
    const float* __restrict__ beta, const float* __restrict__ mean,
    const float* __restrict__ var, float* __restrict__ xT) {
  __shared__ float tile[64][65];
  const int b  = blockIdx.y;
  const int n0 = blockIdx.x * 64;
  const int tx = threadIdx.x & 63;
  const int ty = threadIdx.x >> 6;
  const float* pb = pts + (size_t)b * DFEAT * NPTS;
#pragma unroll
  for (int r = 0; r < 16; ++r) {
    const int d = ty * 16 + r;
    const float s  = gamma[d] / sqrtf(var[d] + BN_EPS);
    const float sh = beta[d] - mean[d] * s;
    float v = fmaf(pb[(size_t)d * NPTS + n0 + tx], s, sh);
    v = (v >= 0.0f) ? v : SLOPE * v;
    tile[d][tx] = v;
  }
  __syncthreads();
  float* xb = xT + ((size_t)b * NPTS + n0) * DFEAT;
#pragma unroll
  for (int r = 0; r < 16; ++r) {
    const int n = ty * 16 + r;
    xb[(size_t)n * DFEAT + tx] = tile[tx][n];
  }
}

// ---------------------------------------------------------------------------
// Neighborhood aggregate: aggT[row][:] = xT[row][:] + sum_j xT[idx[row][j]][:]
// ---------------------------------------------------------------------------
__global__ __launch_bounds__(256) void aggregate_kernel(
    const float* __restrict__ xT, const int* __restrict__ idx,
    float* __restrict__ aggT) {
  const int gid = blockIdx.x * 256 + threadIdx.x;
  const int row = gid >> 4;
  const int l   = gid & 15;
  const int b   = row >> 12;
  const int n   = row & (NPTS - 1);
  const int* ib = idx + (size_t)row * KNN;
  const float4* xb = (const float4*)(xT + (size_t)b * NPTS * DFEAT);
  float4 acc = xb[(size_t)n * 16 + l];
#pragma unroll
  for (int j = 0; j < KNN; ++j) {
    const int m = ib[j];
    const float4 v = xb[(size_t)m * 16 + l];
    acc.x += v.x; acc.y += v.y; acc.z += v.z; acc.w += v.w;
  }
  ((float4*)aggT)[(size_t)row * 16 + l] = acc;
}

// ---------------------------------------------------------------------------
// WMMA fp32 GEMM + fused epilogue (bias, *1/9, residual, next-block BN+act
// written transposed). 256 threads = 8 waves; wave -> (M-tile, 4 N-tiles).
// ---------------------------------------------------------------------------
__global__ __launch_bounds__(256) void gemm_res_kernel(
    const float* __restrict__ aggT, const float* __restrict__ Wt,
    const float* __restrict__ bias, const float* __restrict__ prev,
    float* __restrict__ outP, float* __restrict__ xT,
    const float* __restrict__ g2, const float* __restrict__ b2,
    const float* __restrict__ m2, const float* __restrict__ v2,
    int write_next) {
  __shared__ __align__(16) float sW[64 * 64];
  __shared__ __align__(16) float sA[128 * 64];
  const int b  = blockIdx.y;
  const int n0 = blockIdx.x * 128;
  const int t  = threadIdx.x;

#pragma unroll
  for (int r = 0; r < 4; ++r)
    ((float4*)sW)[t + r * 256] = ((const float4*)Wt)[t + r * 256];
  const float4* ag = (const float4*)(aggT + ((size_t)b * NPTS + n0) * DFEAT);
#pragma unroll
  for (int r = 0; r < 8; ++r)
    ((float4*)sA)[t + r * 256] = ag[t + r * 256];
  __syncthreads();

  const int lane = t & 31;
  const int wave = t >> 5;
  const int hi   = lane >> 4;
  const int l15  = lane & 15;
  const int mt   = wave & 3;
  const int nh   = wave >> 2;
  const int arow = mt * 16 + l15;
  const int obase = mt * 16 + hi * 8;   // this lane's 8 output rows (M)

  // A fragments for the whole K loop, held in registers.
  v2f afr[16];
#pragma unroll
  for (int ks = 0; ks < 16; ++ks) {
    const int off = arow * 64 + ks * 4 + hi * 2;
    afr[ks] = v2f{sW[off], sW[off + 1]};
  }

  // Hoisted per-lane epilogue constants (invariant across the 4 N-tiles):
  // bias, and next-block BN scale/shift for rows obase..obase+7.
  float bias8[8], s8[8], sh8[8];
#pragma unroll
  for (int r = 0; r < 8; ++r) {
    const int o = obase + r;
    bias8[r] = bias[o];
    if (write_next) {
      const float s = g2[o] / sqrtf(v2[o] + BN_EPS);
      s8[r]  = s;
      sh8[r] = b2[o] - m2[o] * s;
    }
  }

  for (int nt0 = 0; nt0 < 4; ++nt0) {
    const int nt   = nh * 4 + nt0;
    const int brow = nt * 16 + l15;
    v8f acc = {0.f, 0.f, 0.f, 0.f, 0.f, 0.f, 0.f, 0.f};
#pragma unroll
    for (int ks = 0; ks < 16; ++ks) {
      const int off = brow * 64 + ks * 4 + hi * 2;
      v2f bfr = v2f{sA[off], sA[off + 1]};
      acc = __builtin_amdgcn_wmma_f32_16x16x4_f32(
          false, afr[ks], false, bfr, (short)0, acc, false, false);
    }
    const int col = n0 + nt * 16 + l15;
    float res[8];
#pragma unroll
    for (int r = 0; r < 8; ++r) {
      const int o = obase + r;
      const float v = fmaf(acc[r], INV9, bias8[r]) +
                      prev[((size_t)b * DFEAT + o) * NPTS + col];
      outP[((size_t)b * DFEAT + o) * NPTS + col] = v;
      res[r] = v;
    }
    if (write_next) {
#pragma unroll
      for (int r = 0; r < 8; ++r) {
        const float x = fmaf(res[r], s8[r], sh8[r]);
        res[r] = (x >= 0.0f) ? x : SLOPE * x;
      }
      float* xrow = xT + ((size_t)b * NPTS + col) * DFEAT + obase;
      float4 w0 = {res[0], res[1], res[2], res[3]};
      float4 w1 = {res[4], res[5], res[6], res[7]};
      ((float4*)xrow)[0] = w0;
      ((float4*)xrow)[1] = w1;
    }
  }
}

// ---------------------------------------------------------------------------
// Launcher. Workspace (33 MB): idx[0,1MB) xT[1,9) aggT[9,17) P0[17,25) P1[25,33)
// ---------------------------------------------------------------------------
extern "C" void kernel_launch(void* const* d_in, const int* in_sizes, int n_in,
                              void* d_out, int out_size, void* d_ws,
                              size_t ws_size, hipStream_t stream) {
  const float* xyz    = (const float*)d_in[0];
  const float* points = (const float*)d_in[1];
  const float* g      = (const float*)d_in[2];
  const float* be     = (const float*)d_in[3];
  const float* mn     = (const float*)d_in[4];
  const float* vr     = (const float*)d_in[5];
  const float* Wc     = (const float*)d_in[6];
  const float* bc     = (const float*)d_in[7];
  float* out = (float*)d_out;

  char* ws = (char*)d_ws;
  int*   idx  = (int*)ws;
  float* xT   = (float*)(ws + (size_t)(1 << 20));
  float* aggT = (float*)(ws + (size_t)(9 << 20));
  float* P0   = (float*)(ws + (size_t)(17 << 20));
  float* P1   = (float*)(ws + (size_t)(25 << 20));

  knn_kernel<<<dim3(NPTS / 256, NBATCH), 256, 0, stream>>>(xyz, idx);
  bnact_transpose_kernel<<<dim3(NPTS / 64, NBATCH), 256, 0, stream>>>(
      points, g, be, mn, vr, xT);

  const float* prev = points;
  for (int i = 0; i < NBLK; ++i) {
    aggregate_kernel<<<dim3(NBATCH * NPTS / 16), 256, 0, stream>>>(xT, idx,
                                                                   aggT);
    float* outP = (i == NBLK - 1) ? out : ((i & 1) ? P1 : P0);
    const int wn = (i < NBLK - 1) ? 1 : 0;
    const size_t nx = (size_t)((i + 1 < NBLK) ? (i + 1) : i) * 64;  // guard
    gemm_res_kernel<<<dim3(NPTS / 128, NBATCH), 256, 0, stream>>>(
        aggT, Wc + (size_t)i * 64 * 64, bc + (size_t)i * 64, prev, outP, xT,
        g + nx, be + nx, mn + nx, vr + nx, wn);
    prev = outP;
  }
}